// TopologicalSignatureDistance_61804579389809
// MI455X (gfx1250) — compile-verified
//
#include <hip/hip_runtime.h>
#include <hip/hip_bf16.h>
#include <stdint.h>

// Problem constants (reference: N=4096 points, two NxN f32 distance matrices).
#define NPTS   4096
#define NEDGE  (NPTS - 1)

#define T1     512           // threads per Prim block (16 waves of 32)
#define K1     (NPTS / T1)   // 8 nodes owned per thread
#define NW1    (T1 / 32)     // 16 waves

#define T2     512           // threads for finalize block
#define NW2    (T2 / 32)

// Pack (float value, node index) so that u64-min == lexicographic
// (value, index) min. Valid because distances are non-negative floats
// (bit pattern monotone as unsigned) — matches jnp.argmin first-min tiebreak.
__device__ __forceinline__ unsigned long long pack_vi(float v, unsigned idx) {
    return ((unsigned long long)__float_as_uint(v) << 32) | (unsigned long long)idx;
}

__device__ __forceinline__ unsigned long long shfl_xor_u64(unsigned long long v, int m) {
    unsigned lo = (unsigned)(v & 0xFFFFFFFFull);
    unsigned hi = (unsigned)(v >> 32);
    lo = __shfl_xor(lo, m, 32);
    hi = __shfl_xor(hi, m, 32);
    return ((unsigned long long)hi << 32) | (unsigned long long)lo;
}

// ---------------------------------------------------------------------------
// Kernel 1: vectorized Prim's MST, one workgroup per matrix (gridDim.x == 2).
// State lives in registers: thread t owns nodes {t + k*512, k=0..7}.
// Per iteration: packed argmin (wave shuffle + LDS cross-wave), owner records
// the MST edge, then row D[j] is DMA'd into LDS with the CDNA5 async
// global->LDS engine and used for the relax step.
// ---------------------------------------------------------------------------
__global__ __launch_bounds__(T1)
void prim_mst_kernel(const float* __restrict__ Da, const float* __restrict__ Db,
                     int2* __restrict__ pa, int2* __restrict__ pb)
{
    const float* __restrict__ D  = (blockIdx.x == 0) ? Da : Db;
    int2* __restrict__ pairs     = (blockIdx.x == 0) ? pa : pb;

    __shared__ __align__(16) float row[NPTS];          // 16 KB staging buffer
    __shared__ unsigned long long wmin[NW1];           // per-wave argmin

    const int t    = threadIdx.x;
    const int wave = t >> 5;
    const int lane = t & 31;
    const float INF = __uint_as_float(0x7F800000u);

    float    md[K1];        // min_dist for owned nodes
    int      par[K1];       // parent for owned nodes
    unsigned intree = 0u;   // in-tree bit per owned slot

    // init: min_dist = D[0,:], min_dist[0] = INF, parent = 0, in_tree = {0}
    #pragma unroll
    for (int k = 0; k < K1; ++k) {
        md[k]  = D[t + k * T1];
        par[k] = 0;
    }
    if (t == 0) { md[0] = INF; intree = 1u; }

    const unsigned long long gD = (unsigned long long)(const void*)D;
    // Low 32 bits of the generic pointer to a __shared__ object are the LDS
    // byte offset (generic = {SHARED_BASE aperture, offset}).
    const unsigned ldsrow = (unsigned)(uintptr_t)(&row[0]);

    for (int it = 0; it < NEDGE; ++it) {
        // ---- local packed argmin over owned nodes (in-tree nodes hold INF) ----
        unsigned long long best = pack_vi(md[0], (unsigned)t);
        #pragma unroll
        for (int k = 1; k < K1; ++k) {
            unsigned long long p = pack_vi(md[k], (unsigned)(t + k * T1));
            best = (p < best) ? p : best;
        }
        // ---- wave32 reduction ----
        #pragma unroll
        for (int m = 16; m >= 1; m >>= 1) {
            unsigned long long o = shfl_xor_u64(best, m);
            best = (o < best) ? o : best;
        }
        if (lane == 0) wmin[wave] = best;
        __syncthreads();

        // ---- cross-wave reduction: every thread computes j itself ----
        unsigned long long b = wmin[0];
        #pragma unroll
        for (int w = 1; w < NW1; ++w) {
            unsigned long long o = wmin[w];
            b = (o < b) ? o : b;
        }
        const int j = (int)(unsigned)(b & 0xFFFFFFFFull);

        // ---- owner of node j records edge (parent[j], j), marks in-tree ----
        if ((j & (T1 - 1)) == t) {
            const int slot = j >> 9;            // j / T1
            pairs[it] = make_int2(par[slot], j);
            md[slot]  = INF;
            intree   |= (1u << slot);
        }

        // ---- async DMA row D[j] (16 KB) into LDS: 32 B per lane ----
        {
            unsigned l = ldsrow + (unsigned)(t * 32);
            unsigned long long g = gD
                + (unsigned long long)(unsigned)j * (unsigned long long)(NPTS * 4)
                + (unsigned long long)(t * 32);
            asm volatile("global_load_async_to_lds_b128 %0, %1, off\n\t"
                         "global_load_async_to_lds_b128 %0, %1, off offset:16"
                         :: "v"(l), "v"(g) : "memory");
            asm volatile("s_wait_asynccnt 0" ::: "memory");
        }
        __syncthreads();  // row visible to all waves; also protects wmin reuse

        // ---- relax: better = (D[j,k] < min_dist[k]) && !in_tree[k] ----
        #pragma unroll
        for (int k = 0; k < K1; ++k) {
            float d = row[t + k * T1];
            bool better = (d < md[k]) && (((intree >> k) & 1u) == 0u);
            if (better) { md[k] = d; par[k] = j; }
        }
    }
}

// ---------------------------------------------------------------------------
// Kernel 2: gather signatures from both matrices along both edge sets,
// reduce squared differences, count ordered-pair intersection, emit
// (distance, distance1_2, distance2_1, matched_pairs).
// ---------------------------------------------------------------------------
__global__ __launch_bounds__(T2)
void finalize_kernel(const float* __restrict__ D1, const float* __restrict__ D2,
                     const int2* __restrict__ pairs1, const int2* __restrict__ pairs2,
                     float* __restrict__ out)
{
    __shared__ unsigned keys2[NEDGE];
    __shared__ float r1[NW2], r2[NW2];
    __shared__ int   rc[NW2];

    const int t = threadIdx.x;

    for (int i = t; i < NEDGE; i += T2) {
        int2 e = pairs2[i];
        keys2[i] = ((unsigned)e.x << 12) | (unsigned)e.y;   // N=4096 -> 12 bits
    }
    __syncthreads();

    float acc1 = 0.f, acc2 = 0.f;
    unsigned mykeys[8];
    #pragma unroll
    for (int q = 0; q < 8; ++q) mykeys[q] = 0xFFFFFFFFu;    // sentinel (>24 bits)

    int nm = 0;
    for (int i = t; i < NEDGE; i += T2) {
        int2 e1 = pairs1[i];
        size_t i1 = (size_t)e1.x * NPTS + (size_t)e1.y;
        float a = D1[i1], bb = D2[i1];
        float d = a - bb;
        acc1 += d * d;

        int2 e2 = pairs2[i];
        size_t i2 = (size_t)e2.x * NPTS + (size_t)e2.y;
        float c = D2[i2], dd = D1[i2];
        float d2 = c - dd;
        acc2 += d2 * d2;

        mykeys[nm++] = ((unsigned)e1.x << 12) | (unsigned)e1.y;
    }

    // intersection: for each of my pairs1 keys, does it occur in pairs2?
    unsigned matchedbits = 0u;
    for (int jj = 0; jj < NEDGE; ++jj) {
        unsigned k2 = keys2[jj];                 // LDS broadcast read
        #pragma unroll
        for (int q = 0; q < 8; ++q)
            if (mykeys[q] == k2) matchedbits |= (1u << q);
    }
    int cnt = __popc(matchedbits);

    // block reduction
    #pragma unroll
    for (int m = 16; m >= 1; m >>= 1) {
        acc1 += __shfl_xor(acc1, m, 32);
        acc2 += __shfl_xor(acc2, m, 32);
        cnt  += __shfl_xor(cnt,  m, 32);
    }
    const int wave = t >> 5, lane = t & 31;
    if (lane == 0) { r1[wave] = acc1; r2[wave] = acc2; rc[wave] = cnt; }
    __syncthreads();
    if (t == 0) {
        float s1 = 0.f, s2 = 0.f; int c = 0;
        for (int w = 0; w < NW2; ++w) { s1 += r1[w]; s2 += r2[w]; c += rc[w]; }
        float d12 = sqrtf(s1), d21 = sqrtf(s2);
        out[0] = d12 + d21;     // distance
        out[1] = d12;           // distance1_2
        out[2] = d21;           // distance2_1
        out[3] = (float)c;      // matched_pairs
    }
}

// ---------------------------------------------------------------------------
extern "C" void kernel_launch(void* const* d_in, const int* in_sizes, int n_in,
                              void* d_out, int out_size, void* d_ws, size_t ws_size,
                              hipStream_t stream)
{
    (void)in_sizes; (void)n_in; (void)out_size; (void)ws_size;

    const float* D1 = (const float*)d_in[0];
    const float* D2 = (const float*)d_in[1];

    int2* pairs1 = (int2*)d_ws;            // NEDGE * 8 bytes
    int2* pairs2 = pairs1 + NEDGE;         // NEDGE * 8 bytes  (total ~64 KB)

    prim_mst_kernel<<<2, T1, 0, stream>>>(D1, D2, pairs1, pairs2);
    finalize_kernel<<<1, T2, 0, stream>>>(D1, D2, pairs1, pairs2, (float*)d_out);
}